// GCN_64123861729517
// MI455X (gfx1250) — compile-verified
//
#include <hip/hip_runtime.h>
#include <hip/hip_bf16.h>

typedef float v2f __attribute__((ext_vector_type(2)));
typedef float v8f __attribute__((ext_vector_type(8)));

// ---------------------------------------------------------------- utilities

__global__ void zero_f(float* __restrict__ p, int n) {
    int i = blockIdx.x * blockDim.x + threadIdx.x;
    int stride = gridDim.x * blockDim.x;
    for (; i < n; i += stride) p[i] = 0.0f;
}

// deg_out[src]++ , deg_in[dst]++
__global__ void degrees_k(const int* __restrict__ src, const int* __restrict__ dst,
                          float* __restrict__ deg_out, float* __restrict__ deg_in, int E) {
    int t = blockIdx.x * blockDim.x + threadIdx.x;
    if (t >= E) return;
    atomicAdd(deg_out + src[t], 1.0f);
    atomicAdd(deg_in + dst[t], 1.0f);
}

// in-place: d = rsqrt(max(d,1))
__global__ void rsqrt_clamp_k(float* __restrict__ d, int n) {
    int i = blockIdx.x * blockDim.x + threadIdx.x;
    if (i >= n) return;
    d[i] = rsqrtf(fmaxf(d[i], 1.0f));
}

// xs[i][0..6] = x[i][0..6] * norm_out[i]; xs[i][7] = 0   (stride-8 padded)
__global__ void prescale_x_k(const float* __restrict__ x, const float* __restrict__ norm_out,
                             float* __restrict__ xs, int N) {
    int i = blockIdx.x * blockDim.x + threadIdx.x;
    if (i >= N) return;
    float no = norm_out[i];
    const float* xr = x + (size_t)i * 7;
    float* o = xs + (size_t)i * 8;
#pragma unroll
    for (int c = 0; c < 7; ++c) o[c] = xr[c] * no;
    o[7] = 0.0f;
}

// ---------------------------------------------------------------- edge scatter

// layer-1: 2 threads/edge, 8-wide features, edge weight
__global__ void edge_agg8_k(const float* __restrict__ xs, const float* __restrict__ ew,
                            const int* __restrict__ src, const int* __restrict__ dst,
                            float* __restrict__ agg, int E) {
    int t = blockIdx.x * blockDim.x + threadIdx.x;
    if (t >= E * 2) return;
    int e = t >> 1, j = (t & 1) * 4;
    int s = src[e], d = dst[e];
    float w = ew[e];
    const float4 v = *(const float4*)(xs + (size_t)s * 8 + j);
    float* p = agg + (size_t)d * 8 + j;
    atomicAdd(p + 0, v.x * w);
    atomicAdd(p + 1, v.y * w);
    atomicAdd(p + 2, v.z * w);
    atomicAdd(p + 3, v.w * w);
}

// layers 2-4: 8 threads/edge, 32-wide features (hs already scaled by norm_out)
__global__ void edge_agg32_k(const float* __restrict__ hs,
                             const int* __restrict__ src, const int* __restrict__ dst,
                             float* __restrict__ agg, int E) {
    int t = blockIdx.x * blockDim.x + threadIdx.x;
    if (t >= E * 8) return;
    int e = t >> 3, j = (t & 7) * 4;
    int s = src[e], d = dst[e];
    const float4 v = *(const float4*)(hs + (size_t)s * 32 + j);
    float* p = agg + (size_t)d * 32 + j;
    atomicAdd(p + 0, v.x);
    atomicAdd(p + 1, v.y);
    atomicAdd(p + 2, v.z);
    atomicAdd(p + 3, v.w);
}

// ---------------------------------------------------------------- WMMA GEMM
// One wave -> 16 rows x 32 cols. out = relu((agg*norm_in) @ W + b) * norm_out
// K = 8 (layer 1, padded) or 32. Wrows = valid rows of W in memory (7 or 32).
template <int K>
__global__ void gcn_gemm_wmma_k(const float* __restrict__ agg,
                                const float* __restrict__ norm_in,
                                const float* __restrict__ norm_out,
                                const float* __restrict__ W,   // [Wrows,32] row-major
                                const float* __restrict__ bias,// [32]
                                float* __restrict__ out,       // [N,32]
                                int N, int Wrows) {
    __shared__ float Wlds[K * 32];
    for (int idx = threadIdx.x; idx < K * 32; idx += blockDim.x) {
        int r = idx >> 5;
        Wlds[idx] = (r < Wrows) ? W[r * 32 + (idx & 31)] : 0.0f;
    }
    __syncthreads();

    const int wave = threadIdx.x >> 5;
    const int lane = threadIdx.x & 31;
    const int hi   = lane >> 4;      // half-wave selects K pair
    const int lx   = lane & 15;
    const int rowbase = (blockIdx.x * 8 + wave) * 16;

    int rowA = rowbase + lx;
    int rA = rowA < N ? rowA : (N - 1);     // clamp; EXEC must stay all-1 for WMMA
    const float ni = norm_in[rA];

    v8f c0 = {0.f, 0.f, 0.f, 0.f, 0.f, 0.f, 0.f, 0.f};
    v8f c1 = {0.f, 0.f, 0.f, 0.f, 0.f, 0.f, 0.f, 0.f};

#pragma unroll
    for (int kk = 0; kk < K / 4; ++kk) {
        const int kbase = kk * 4 + 2 * hi;
        const float2 av = *(const float2*)(agg + (size_t)rA * K + kbase);
        v2f a; a.x = av.x * ni; a.y = av.y * ni;
        v2f b0, b1;
        b0.x = Wlds[kbase * 32 + lx];
        b0.y = Wlds[(kbase + 1) * 32 + lx];
        b1.x = Wlds[kbase * 32 + 16 + lx];
        b1.y = Wlds[(kbase + 1) * 32 + 16 + lx];
        c0 = __builtin_amdgcn_wmma_f32_16x16x4_f32(false, a, false, b0, (short)0, c0, false, false);
        c1 = __builtin_amdgcn_wmma_f32_16x16x4_f32(false, a, false, b1, (short)0, c1, false, false);
    }

    const float bias0 = bias[lx];
    const float bias1 = bias[16 + lx];
#pragma unroll
    for (int v = 0; v < 8; ++v) {
        int row = rowbase + v + 8 * hi;     // C/D layout: VGPR v -> M = v (+8 for hi half)
        if (row < N) {
            float no = norm_out[row];
            out[(size_t)row * 32 + lx]      = fmaxf(c0[v] + bias0, 0.0f) * no;
            out[(size_t)row * 32 + 16 + lx] = fmaxf(c1[v] + bias1, 0.0f) * no;
        }
    }
}

// ---------------------------------------------------------------- layer 4 (32 -> 2)

__global__ void layer4_k(const float* __restrict__ agg, const float* __restrict__ norm_in,
                         const float* __restrict__ W4, const float* __restrict__ b4,
                         float* __restrict__ h4, int N) {
    int i = blockIdx.x * blockDim.x + threadIdx.x;
    if (i >= N) return;
    float ni = norm_in[i];
    float a0 = b4[0], a1 = b4[1];
    const float* r = agg + (size_t)i * 32;
#pragma unroll
    for (int k = 0; k < 32; ++k) {
        float a = r[k] * ni;
        a0 += a * W4[k * 2 + 0];
        a1 += a * W4[k * 2 + 1];
    }
    h4[(size_t)i * 2 + 0] = a0;
    h4[(size_t)i * 2 + 1] = a1;
}

// ---------------------------------------------------------------- per-graph mean

__global__ void readout_acc_k(const float* __restrict__ h4, const int* __restrict__ gid,
                              float* __restrict__ gsum, float* __restrict__ gcnt, int N) {
    int i = blockIdx.x * blockDim.x + threadIdx.x;
    if (i >= N) return;
    int g = gid[i];
    atomicAdd(gsum + (size_t)g * 2 + 0, h4[(size_t)i * 2 + 0]);
    atomicAdd(gsum + (size_t)g * 2 + 1, h4[(size_t)i * 2 + 1]);
    atomicAdd(gcnt + g, 1.0f);
}

__global__ void readout_div_k(const float* __restrict__ gsum, const float* __restrict__ gcnt,
                              float* __restrict__ out, int G) {
    int g = blockIdx.x * blockDim.x + threadIdx.x;
    if (g >= G) return;
    float c = fmaxf(gcnt[g], 1.0f);
    out[g * 2 + 0] = gsum[g * 2 + 0] / c;
    out[g * 2 + 1] = gsum[g * 2 + 1] / c;
}

// ---------------------------------------------------------------- launch

extern "C" void kernel_launch(void* const* d_in, const int* in_sizes, int n_in,
                              void* d_out, int out_size, void* d_ws, size_t ws_size,
                              hipStream_t stream) {
    const float* x   = (const float*)d_in[0];
    const float* ew  = (const float*)d_in[1];
    const float* W1  = (const float*)d_in[2];
    const float* b1  = (const float*)d_in[3];
    const float* W2  = (const float*)d_in[4];
    const float* b2  = (const float*)d_in[5];
    const float* W3  = (const float*)d_in[6];
    const float* b3  = (const float*)d_in[7];
    const float* W4  = (const float*)d_in[8];
    const float* b4  = (const float*)d_in[9];
    const int*   src = (const int*)d_in[10];
    const int*   dst = (const int*)d_in[11];
    const int*   gid = (const int*)d_in[12];

    const int N = in_sizes[12];
    const int E = in_sizes[1];
    const int G = out_size / 2;
    float* out = (float*)d_out;

    // workspace layout (floats)
    float* ws       = (float*)d_ws;
    float* norm_out = ws;                       // N
    float* norm_in  = ws + (size_t)N;           // N
    float* xs       = ws + (size_t)2 * N;       // 8N
    float* agg      = ws + (size_t)10 * N;      // 32N
    float* hs       = ws + (size_t)42 * N;      // 32N
    float* h4       = ws + (size_t)74 * N;      // 2N
    float* gsum     = ws + (size_t)76 * N;      // 2G
    float* gcnt     = gsum + (size_t)2 * G;     // G

    const int B = 256;
    const int gN   = (N + B - 1) / B;
    const int gE   = (E + B - 1) / B;
    const int gE2  = (E * 2 + B - 1) / B;
    const int gE8  = (E * 8 + B - 1) / B;
    const int gGem = (N + 127) / 128;           // 8 waves x 16 rows

    // degrees -> norms (in place)
    zero_f<<<1024, B, 0, stream>>>(norm_out, 2 * N);
    degrees_k<<<gE, B, 0, stream>>>(src, dst, norm_out, norm_in, E);
    rsqrt_clamp_k<<<(2 * N + B - 1) / B, B, 0, stream>>>(norm_out, 2 * N);

    // layer 1: xs = x * norm_out (padded K=8); agg8 = scatter(xs[src]*e); hs = relu(...)
    prescale_x_k<<<gN, B, 0, stream>>>(x, norm_out, xs, N);
    zero_f<<<1024, B, 0, stream>>>(agg, 8 * N);
    edge_agg8_k<<<gE2, B, 0, stream>>>(xs, ew, src, dst, agg, E);
    gcn_gemm_wmma_k<8><<<gGem, B, 0, stream>>>(agg, norm_in, norm_out, W1, b1, hs, N, 7);

    // layer 2
    zero_f<<<2048, B, 0, stream>>>(agg, 32 * N);
    edge_agg32_k<<<gE8, B, 0, stream>>>(hs, src, dst, agg, E);
    gcn_gemm_wmma_k<32><<<gGem, B, 0, stream>>>(agg, norm_in, norm_out, W2, b2, hs, N, 32);

    // layer 3
    zero_f<<<2048, B, 0, stream>>>(agg, 32 * N);
    edge_agg32_k<<<gE8, B, 0, stream>>>(hs, src, dst, agg, E);
    gcn_gemm_wmma_k<32><<<gGem, B, 0, stream>>>(agg, norm_in, norm_out, W3, b3, hs, N, 32);

    // layer 4 (32 -> 2, no relu, no norm_out on output)
    zero_f<<<2048, B, 0, stream>>>(agg, 32 * N);
    edge_agg32_k<<<gE8, B, 0, stream>>>(hs, src, dst, agg, E);
    layer4_k<<<gN, B, 0, stream>>>(agg, norm_in, W4, b4, h4, N);

    // per-graph mean
    zero_f<<<16, B, 0, stream>>>(gsum, 3 * G);
    readout_acc_k<<<gN, B, 0, stream>>>(h4, gid, gsum, gcnt, N);
    readout_div_k<<<(G + B - 1) / B, B, 0, stream>>>(gsum, gcnt, out, G);
}